// GlobalPointCloudOptimizer_37237366456381
// MI455X (gfx1250) — compile-verified
//
#include <hip/hip_runtime.h>
#include <hip/hip_bf16.h>

// ---------------- problem constants (from reference) ----------------
#define VNUM 24
#define KNUM 4
#define PNUM 96                  // V*K
#define HNUM 224
#define WNUM 224
#define HWN  (HNUM * WNUM)       // 50176 (multiple of 4)
#define NSEG 32                  // segments per pair for avg_dist pass
#define MSEG 32                  // segments per pair for loss pass
#define MCHUNK (HWN / MSEG)      // 1568 pixels per segment

// group-of-4-points view (float4-vectorized streaming)
#define Q_PER_PAIR (HWN * 3 / 4)         // 37632 float4 per pts3d pair-row
#define GRP_I      (HWN / 4)             // 12544 groups in pts_i
#define GRP_PER_PAIR (2 * HWN / 4)       // 25088 groups (i then j)
#define GRP_PER_SEG  (GRP_PER_PAIR / NSEG) // 784 groups per avg segment
#define MGRP       (MCHUNK / 4)          // 392 pixel-groups per loss segment

// workspace layout (float offsets)
#define WS_AVG_OFF   0                   // P*NSEG = 3072 floats
#define WS_FACT_OFF  4096                // P floats
#define WS_LOSS_OFF  8192                // P*MSEG = 3072 floats

typedef float v2f __attribute__((ext_vector_type(2)));
typedef float v8f __attribute__((ext_vector_type(8)));

// ---------------- WMMA-based wave32 reduction ----------------
// A (16x4 f32, 2 VGPRs): v0 = s, v1 = 0 => A[m][0]=s_m, A[m][2]=s_{m+16}.
// B = all-ones => D[m][n] = s_m + s_{m+16} for every n (layout-independent).
// Sum the 8 D regs (uniform per 16-lane half), shfl_xor(16) completes the wave.
__device__ __forceinline__ float wave_reduce_wmma(float s) {
  v2f a; a[0] = s;    a[1] = 0.0f;
  v2f b; b[0] = 1.0f; b[1] = 1.0f;
  v8f c = {0.f, 0.f, 0.f, 0.f, 0.f, 0.f, 0.f, 0.f};
  v8f d = __builtin_amdgcn_wmma_f32_16x16x4_f32(
      /*neg_a=*/false, a, /*neg_b=*/false, b,
      /*c_mod=*/(short)0, c, /*reuse_a=*/false, /*reuse_b=*/false);
  float t = ((d[0] + d[1]) + (d[2] + d[3])) + ((d[4] + d[5]) + (d[6] + d[7]));
  t += __shfl_xor(t, 16, 32);
  return t;  // full 32-lane sum, uniform across the wave
}

// deterministic block reduction (result valid in wave 0)
__device__ __forceinline__ float block_reduce_wmma(float s, float* smem) {
  const int lane = threadIdx.x & 31;
  const int wid  = threadIdx.x >> 5;
  float w = wave_reduce_wmma(s);
  if (lane == 0) smem[wid] = w;
  __syncthreads();
  float r = 0.0f;
  if (wid == 0) {
    const int nw = blockDim.x >> 5;
    float v = (lane < nw) ? smem[lane] : 0.0f;
    r = wave_reduce_wmma(v);
  }
  return r;
}

// unpack a group of 4 xyz points from three float4s
__device__ __forceinline__ void unpack4(const float4& A, const float4& B,
                                        const float4& C,
                                        float (&x)[4], float (&y)[4], float (&z)[4]) {
  x[0] = A.x; y[0] = A.y; z[0] = A.z;
  x[1] = A.w; y[1] = B.x; z[1] = B.y;
  x[2] = B.z; y[2] = B.w; z[2] = C.x;
  x[3] = C.y; y[3] = C.z; z[3] = C.w;
}

// ---------------- kernel 1: per-pair norm partial sums ----------------
// grid = P*NSEG blocks of 256; b128 loads, 4 points / thread / iteration
__global__ void k_avg_partials(const float4* __restrict__ pi4,
                               const float4* __restrict__ pj4,
                               float* __restrict__ partials) {
  __shared__ float smem[8];
  const int p   = blockIdx.x / NSEG;
  const int seg = blockIdx.x % NSEG;
  const int baseq = p * Q_PER_PAIR;
  const int g0 = seg * GRP_PER_SEG;
  float acc = 0.0f;
  for (int g = g0 + threadIdx.x; g < g0 + GRP_PER_SEG; g += blockDim.x) {
    const float4* src = (g < GRP_I) ? (pi4 + baseq + g * 3)
                                    : (pj4 + baseq + (g - GRP_I) * 3);
    __builtin_prefetch(src + 3 * 256, 0, 3);  // global_prefetch_b8
    const float4 A = src[0], B = src[1], C = src[2];
    float x[4], y[4], z[4];
    unpack4(A, B, C, x, y, z);
#pragma unroll
    for (int k = 0; k < 4; ++k)
      acc += sqrtf(x[k] * x[k] + y[k] * y[k] + z[k] * z[k]);
  }
  const float r = block_reduce_wmma(acc, smem);
  if (threadIdx.x == 0) partials[blockIdx.x] = r;
}

// ---------------- kernel 2: fold partials -> factor[p] ----------------
__global__ void k_factors(const float* __restrict__ partials,
                          const float* __restrict__ scales_log,
                          float* __restrict__ factor) {
  const int p = threadIdx.x;
  if (p < PNUM) {
    float s = 0.0f;
    for (int i = 0; i < NSEG; ++i) s += partials[p * NSEG + i];
    const float avg = s * (1.0f / (float)(2 * HWN));
    factor[p] = expf(scales_log[p]) / avg;   // exp(scale_log) / avg_dist
  }
}

// ---------------- kernel 3: weighted-norm loss partials ----------------
// grid = P*MSEG blocks of 256; p uniform per block; 4 pixels / thread / iter
__global__ void k_loss(const float4* __restrict__ pi4,
                       const float4* __restrict__ pj4,
                       const float4* __restrict__ ci4,
                       const float4* __restrict__ cj4,
                       const float*  __restrict__ poses,     // V*16
                       const float*  __restrict__ focals,    // V*2
                       const float*  __restrict__ pp,        // V*2
                       const float4* __restrict__ dlog4,     // V*HW/4
                       const int*    __restrict__ v1idx,
                       const int*    __restrict__ v2idx,
                       const float*  __restrict__ factor,
                       float* __restrict__ loss_partials) {
  __shared__ float sR[VNUM][9];
  __shared__ float sT[VNUM][3];
  __shared__ float sC[VNUM][4];  // fx, fy, ppx, ppy
  __shared__ float smem[8];

  for (int v = threadIdx.x; v < VNUM; v += blockDim.x) {
    const float* Pv = poses + v * 16;
    sR[v][0] = Pv[0];  sR[v][1] = Pv[1];  sR[v][2] = Pv[2];  sT[v][0] = Pv[3];
    sR[v][3] = Pv[4];  sR[v][4] = Pv[5];  sR[v][5] = Pv[6];  sT[v][1] = Pv[7];
    sR[v][6] = Pv[8];  sR[v][7] = Pv[9];  sR[v][8] = Pv[10]; sT[v][2] = Pv[11];
    sC[v][0] = focals[v * 2];  sC[v][1] = focals[v * 2 + 1];
    sC[v][2] = pp[v * 2];      sC[v][3] = pp[v * 2 + 1];
  }
  __syncthreads();

  const int p   = blockIdx.x / MSEG;
  const int seg = blockIdx.x % MSEG;
  const int v1  = v1idx[p];
  const int v2  = v2idx[p];
  const float f = factor[p];

  const float r10 = sR[v1][0], r11 = sR[v1][1], r12 = sR[v1][2];
  const float r13 = sR[v1][3], r14 = sR[v1][4], r15 = sR[v1][5];
  const float r16 = sR[v1][6], r17 = sR[v1][7], r18 = sR[v1][8];
  const float t10 = sT[v1][0], t11 = sT[v1][1], t12 = sT[v1][2];
  const float r20 = sR[v2][0], r21 = sR[v2][1], r22 = sR[v2][2];
  const float r23 = sR[v2][3], r24 = sR[v2][4], r25 = sR[v2][5];
  const float r26 = sR[v2][6], r27 = sR[v2][7], r28 = sR[v2][8];
  const float t20 = sT[v2][0], t21 = sT[v2][1], t22 = sT[v2][2];
  const float fx1 = sC[v1][0], fy1 = sC[v1][1], px1 = sC[v1][2], py1 = sC[v1][3];
  const float fx2 = sC[v2][0], fy2 = sC[v2][1], px2 = sC[v2][2], py2 = sC[v2][3];
  const float ifx1 = 1.0f / fx1, ify1 = 1.0f / fy1;
  const float ifx2 = 1.0f / fx2, ify2 = 1.0f / fy2;

  const int pbaseq = p * Q_PER_PAIR;        // float4 base of pts streams
  const int cbase4 = p * (HWN / 4);         // float4 base of conf streams
  const int d1base4 = v1 * (HWN / 4);
  const int d2base4 = v2 * (HWN / 4);
  const int g0 = seg * MGRP;
  float acc = 0.0f;

  for (int g = g0 + threadIdx.x; g < g0 + MGRP; g += blockDim.x) {
    const int m = g * 4;                    // first pixel of this group
    const float4* si = pi4 + pbaseq + g * 3;
    const float4* sj = pj4 + pbaseq + g * 3;
    __builtin_prefetch(si + 3 * 256, 0, 3);
    __builtin_prefetch(sj + 3 * 256, 0, 3);
    const float4 Ai = si[0], Bi = si[1], Ci = si[2];
    const float4 Aj = sj[0], Bj = sj[1], Cj = sj[2];
    const float4 ci = ci4[cbase4 + g];
    const float4 cj = cj4[cbase4 + g];
    const float4 q1 = dlog4[d1base4 + g];
    const float4 q2 = dlog4[d2base4 + g];

    float xi[4], yi[4], zi[4], xj[4], yj[4], zj[4];
    unpack4(Ai, Bi, Ci, xi, yi, zi);
    unpack4(Aj, Bj, Cj, xj, yj, zj);
    const float wiv[4] = {logf(ci.x), logf(ci.y), logf(ci.z), logf(ci.w)};
    const float wjv[4] = {logf(cj.x), logf(cj.y), logf(cj.z), logf(cj.w)};
    const float d1v[4] = {expf(q1.x), expf(q1.y), expf(q1.z), expf(q1.w)};
    const float d2v[4] = {expf(q2.x), expf(q2.y), expf(q2.z), expf(q2.w)};

    // 224 % 4 == 0 and m % 4 == 0 => the 4-pixel group stays in one row
    const float py = (float)(m / WNUM);
    const float px0 = (float)(m % WNUM);

#pragma unroll
    for (int k = 0; k < 4; ++k) {
      const float px = px0 + (float)k;
      // pts3d_ours for view1
      const float d1  = d1v[k];
      const float c1x = d1 * (px - px1) * ifx1;
      const float c1y = d1 * (py - py1) * ify1;
      const float o1x = r10 * c1x + r11 * c1y + r12 * d1 + t10;
      const float o1y = r13 * c1x + r14 * c1y + r15 * d1 + t11;
      const float o1z = r16 * c1x + r17 * c1y + r18 * d1 + t12;
      // pts3d_ours for view2
      const float d2  = d2v[k];
      const float c2x = d2 * (px - px2) * ifx2;
      const float c2y = d2 * (py - py2) * ify2;
      const float o2x = r20 * c2x + r21 * c2y + r22 * d2 + t20;
      const float o2y = r23 * c2x + r24 * c2y + r25 * d2 + t21;
      const float o2z = r26 * c2x + r27 * c2y + r28 * d2 + t22;
      // predicted points: T1 applied to f * (normalized pts)
      const float sxi = f * xi[k], syi = f * yi[k], szi = f * zi[k];
      const float pix = r10 * sxi + r11 * syi + r12 * szi + t10;
      const float piy = r13 * sxi + r14 * syi + r15 * szi + t11;
      const float piz = r16 * sxi + r17 * syi + r18 * szi + t12;
      const float sxj = f * xj[k], syj = f * yj[k], szj = f * zj[k];
      const float pjx = r10 * sxj + r11 * syj + r12 * szj + t10;
      const float pjy = r13 * sxj + r14 * syj + r15 * szj + t11;
      const float pjz = r16 * sxj + r17 * syj + r18 * szj + t12;

      const float dix = o1x - pix, diy = o1y - piy, diz = o1z - piz;
      const float djx = o2x - pjx, djy = o2y - pjy, djz = o2z - pjz;
      acc += sqrtf(dix * dix + diy * diy + diz * diz) * wiv[k]
           + sqrtf(djx * djx + djy * djy + djz * djz) * wjv[k];
    }
  }

  const float r = block_reduce_wmma(acc, smem);
  if (threadIdx.x == 0) loss_partials[blockIdx.x] = r;
}

// ---------------- kernel 4: finalize scalar ----------------
__global__ void k_finalize(const float* __restrict__ loss_partials, int n,
                           float* __restrict__ out) {
  __shared__ float smem[8];
  float acc = 0.0f;
  for (int i = threadIdx.x; i < n; i += blockDim.x) acc += loss_partials[i];
  const float r = block_reduce_wmma(acc, smem);
  if (threadIdx.x == 0) out[0] = r * (1.0f / ((float)PNUM * (float)HWN));
}

// ---------------- launcher ----------------
extern "C" void kernel_launch(void* const* d_in, const int* in_sizes, int n_in,
                              void* d_out, int out_size, void* d_ws, size_t ws_size,
                              hipStream_t stream) {
  const float4* pi4   = (const float4*)d_in[0];
  const float4* pj4   = (const float4*)d_in[1];
  const float4* ci4   = (const float4*)d_in[2];
  const float4* cj4   = (const float4*)d_in[3];
  const float* poses      = (const float*)d_in[4];
  const float* focals     = (const float*)d_in[5];
  const float* pp         = (const float*)d_in[6];
  // d_in[7] = pixels : recomputed on the fly from m
  const float4* dlog4     = (const float4*)d_in[8];
  const float* scales_log = (const float*)d_in[9];
  const int*   v1idx      = (const int*)d_in[10];
  const int*   v2idx      = (const int*)d_in[11];
  float* out = (float*)d_out;
  float* ws  = (float*)d_ws;

  float* avg_part  = ws + WS_AVG_OFF;
  float* factor    = ws + WS_FACT_OFF;
  float* loss_part = ws + WS_LOSS_OFF;

  k_avg_partials<<<PNUM * NSEG, 256, 0, stream>>>(pi4, pj4, avg_part);
  k_factors<<<1, 128, 0, stream>>>(avg_part, scales_log, factor);
  k_loss<<<PNUM * MSEG, 256, 0, stream>>>(pi4, pj4, ci4, cj4,
                                          poses, focals, pp, dlog4,
                                          v1idx, v2idx, factor, loss_part);
  k_finalize<<<1, 256, 0, stream>>>(loss_part, PNUM * MSEG, out);
}